// LIFNode_27462020890902
// MI455X (gfx1250) — compile-verified
//
#include <hip/hip_runtime.h>

// LIF spiking recurrence over x[T=32, B=256, D=4096] f32.
// Memory-bound streaming kernel: 128 MiB in + 128 MiB out -> ~11.5us floor @ 23.3 TB/s.
// gfx1250 path used: GLOBAL_LOAD_ASYNC_TO_LDS_B128 + s_wait_asynccnt pipeline,
// b128 vector accesses, non-temporal stores. (No WMMA: zero matmul content.)

typedef float v4f __attribute__((ext_vector_type(4)));
typedef int   v4i __attribute__((ext_vector_type(4)));

#define AS_GLOBAL __attribute__((address_space(1)))
#define AS_LDS    __attribute__((address_space(3)))

#define T_STEPS 32
#define PIPE_DEPTH 4
#define BLOCK 256
#define DECAY 0.25f
#define THRESH 0.5f

#if defined(__AMDGCN__) && __has_builtin(__builtin_amdgcn_global_load_async_to_lds_b128)
#define USE_ASYNC 1
#else
#define USE_ASYNC 0
#endif

#if USE_ASYNC
__device__ __forceinline__ void async_ld_b128(const v4f* g, v4f* l) {
    // Builtin signature (from hipcc diagnostic): param0 = int4 __device__*  (AS1),
    // param1 = int4 LDS*, then imm offset, imm cpol.
    __builtin_amdgcn_global_load_async_to_lds_b128(
        (AS_GLOBAL v4i*)g,
        (AS_LDS v4i*)l,
        0, 0);
}
#endif

__device__ __forceinline__ void lif_step(v4f& mem, v4f& gate, const v4f xt, v4f& spk) {
    mem = mem * gate + xt;                 // v_fma per lane component
#pragma unroll
    for (int k = 0; k < 4; ++k) {
        const bool fire = mem[k] > THRESH; // one v_cmp, two v_cndmask reuse the VCC
        spk[k]  = fire ? 1.0f : 0.0f;
        gate[k] = fire ? 0.0f : DECAY;     // decay * (1 - spike), spike in {0,1}
    }
}

__global__ __launch_bounds__(BLOCK) void lif_forward_kernel(const float* __restrict__ x,
                                                            float* __restrict__ out,
                                                            int n4) {
    const int tid = threadIdx.x;
    const int v   = blockIdx.x * BLOCK + tid;
    if (v >= n4) return;

    const v4f* __restrict__ xv = (const v4f*)x;
    v4f* __restrict__ ov       = (v4f*)out;
    const size_t stride = (size_t)n4;      // v4f elements per timestep

    v4f mem = {0.f, 0.f, 0.f, 0.f};        // mem*gate == 0 at t=0 -> mem = x0
    v4f gate = {0.f, 0.f, 0.f, 0.f};
    v4f spk;

#if USE_ASYNC
    __shared__ v4f stage[PIPE_DEPTH][BLOCK];

    // Prologue: fill the async pipeline (ASYNCcnt tracks these, not VGPR deps).
#pragma unroll
    for (int p = 0; p < PIPE_DEPTH; ++p) {
        async_ld_b128(xv + (size_t)p * stride + v, &stage[p][tid]);
    }

    // Steady state: wait for the head of the pipe, consume from LDS, refill slot.
#pragma unroll
    for (int t = 0; t < T_STEPS - PIPE_DEPTH; ++t) {
        asm volatile("s_wait_asynccnt %0" ::"n"(PIPE_DEPTH - 1) : "memory");
        const v4f xt = stage[t % PIPE_DEPTH][tid];
        lif_step(mem, gate, xt, spk);
        // Slot reuse WAR guard: async LDS writes are unordered vs DS reads,
        // so make sure our ds_load has fully completed before re-targeting the slot.
        asm volatile("s_wait_dscnt 0" ::: "memory");
        async_ld_b128(xv + (size_t)(t + PIPE_DEPTH) * stride + v,
                      &stage[t % PIPE_DEPTH][tid]);
        __builtin_nontemporal_store(spk, ov + (size_t)t * stride + v);
    }

    // Tail: drain the pipeline.
#pragma unroll
    for (int t = T_STEPS - PIPE_DEPTH; t < T_STEPS; ++t) {
        asm volatile("s_wait_asynccnt 0" ::: "memory");
        const v4f xt = stage[t % PIPE_DEPTH][tid];
        lif_step(mem, gate, xt, spk);
        __builtin_nontemporal_store(spk, ov + (size_t)t * stride + v);
    }
#else
    // Fallback: pure-register pipeline with non-temporal b128 loads; the
    // scheduler hoists loads (independent of the serial mem chain) across
    // the unroll window.
#pragma unroll 8
    for (int t = 0; t < T_STEPS; ++t) {
        const v4f xt = __builtin_nontemporal_load(xv + (size_t)t * stride + v);
        lif_step(mem, gate, xt, spk);
        __builtin_nontemporal_store(spk, ov + (size_t)t * stride + v);
    }
#endif
}

extern "C" void kernel_launch(void* const* d_in, const int* in_sizes, int n_in,
                              void* d_out, int out_size, void* d_ws, size_t ws_size,
                              hipStream_t stream) {
    const float* x = (const float*)d_in[0];
    float* out     = (float*)d_out;

    const int total = in_sizes[0];          // T*B*D = 32*256*4096
    const int n     = total / T_STEPS;      // elements per timestep (B*D)
    const int n4    = n / 4;                // v4f vectors per timestep (divisible: D=4096)

    const int blocks = (n4 + BLOCK - 1) / BLOCK;
    lif_forward_kernel<<<blocks, BLOCK, 0, stream>>>(x, out, n4);
}